// Repulsive_89653147337010
// MI455X (gfx1250) — compile-verified
//
#include <hip/hip_runtime.h>
#include <hip/hip_bf16.h>
#include <stdint.h>

// ---------------------------------------------------------------------------
// Repulsive-contact flag kernel for MI455X (gfx1250).
// out[b] = (eng[b] >= sum_a eng_atm[elm[b,a]]) || any pair in structure b with
//          (radius[ei]+radius[ej])^2 >= |pos[n,j]-pos[n,i]|^2
//
// Roofline: n/i/j streams (~100 MB) dominate -> ~4.3 us at 23.3 TB/s.
// pos (1.5 MB) / elm (0.5 MB) gathers are L2-resident. CDNA5 paths used:
//   * global_load_async_to_lds_b128 + s_wait_asynccnt double buffering
//     (in-flight tiles cost no VGPRs; per-lane-private slices -> no barriers)
//   * wave32 shfl_xor OR-reduction of a 64-bit contact bitmask
//   * precomputed (r_i+r_j)^2 table in LDS: 1 gather + 1 compare per pair
// ---------------------------------------------------------------------------

#define BLOCK 256
#define PPT   4                  // pairs per thread per tile (one int4 each)
#define TILE  (BLOCK * PPT)      // 1024 pairs per tile
#define TPB   8                  // tiles per block

__device__ __forceinline__ void async_b128(uint32_t lds_addr, const void* gaddr) {
    // CDNA5: GLOBAL_LOAD_ASYNC_TO_LDS_B128 (tracked with ASYNCcnt).
    asm volatile("global_load_async_to_lds_b128 %0, %1, off"
                 :: "v"(lds_addr), "v"(gaddr) : "memory");
}

__device__ __forceinline__ uint32_t lds_lo32(const void* p) {
    // Generic pointer to __shared__: low 32 bits are the LDS byte offset.
    return (uint32_t)(size_t)p;
}

// ---- Kernel 1: per-structure energy check, also initializes out[] ---------
__global__ void eng_kernel(const float* __restrict__ eng,
                           const int*   __restrict__ elm,
                           const float* __restrict__ eng_atm,
                           float*       __restrict__ out,
                           int A, int E) {
    __shared__ float s_ea[64];
    __shared__ float s_red[BLOCK];
    const int b   = blockIdx.x;
    const int tid = threadIdx.x;
    for (int k = tid; k < E; k += BLOCK) s_ea[k] = eng_atm[k];
    __syncthreads();

    const int* e = elm + (size_t)b * (size_t)A;
    float partial = 0.0f;
    for (int a = tid; a < A; a += BLOCK) partial += s_ea[e[a]];
    s_red[tid] = partial;
    __syncthreads();
    for (int s = BLOCK / 2; s > 0; s >>= 1) {
        if (tid < s) s_red[tid] += s_red[tid + s];
        __syncthreads();
    }
    if (tid == 0) out[b] = (eng[b] >= s_red[0]) ? 1.0f : 0.0f;
}

// ---- Pair predicate: accumulate into a 64-bit per-thread bitmask ----------
__device__ __forceinline__ void do_pair(int nn, int ii, int jj, int A, int E,
                                        const float* __restrict__ pos,
                                        const int*   __restrict__ elm,
                                        const float* s_rs2,
                                        unsigned long long& mask) {
    const int base = nn * A;
    const float* pi = pos + (size_t)(base + ii) * 3u;
    const float* pj = pos + (size_t)(base + jj) * 3u;
    const float dx = pj[0] - pi[0];
    const float dy = pj[1] - pi[1];
    const float dz = pj[2] - pi[2];
    const float sod = dx * dx + dy * dy + dz * dz;
    const float r2  = s_rs2[elm[base + ii] * E + elm[base + jj]];
    const unsigned long long pred = (r2 >= sod) ? 1ull : 0ull;
    mask |= pred << nn;            // B <= 64 structures
}

// ---- Kernel 2: pair stream with async double-buffered LDS tiles -----------
__global__ void __launch_bounds__(BLOCK)
pair_kernel(const float* __restrict__ pos,
            const int*   __restrict__ elm,
            const float* __restrict__ radius,
            const int*   __restrict__ n,
            const int*   __restrict__ i,
            const int*   __restrict__ j,
            float*       __restrict__ out,
            int A, int E, int numTiles, int B) {
    __shared__ int4  s_n[2][BLOCK];
    __shared__ int4  s_i[2][BLOCK];
    __shared__ int4  s_j[2][BLOCK];
    __shared__ float s_rs2[64];                 // (radius[a]+radius[b])^2, E*E<=64
    __shared__ unsigned long long s_mask;

    const int tid = threadIdx.x;
    if (tid == 0) s_mask = 0ull;
    for (int k = tid; k < E * E; k += BLOCK) {
        const float rs = radius[k / E] + radius[k % E];
        s_rs2[k] = rs * rs;
    }
    __syncthreads();

    const int tile0 = blockIdx.x * TPB;
    int ntiles = numTiles - tile0;
    if (ntiles > TPB) ntiles = TPB;

    const int4* np = (const int4*)n;
    const int4* ip = (const int4*)i;
    const int4* jp = (const int4*)j;

    unsigned long long mask = 0ull;

    if (ntiles > 0) {
        // Prologue: issue tile 0 into buffer 0 (3 async b128 ops per lane).
        {
            const size_t off = (size_t)tile0 * BLOCK + tid;
            async_b128(lds_lo32(&s_n[0][tid]), np + off);
            async_b128(lds_lo32(&s_i[0][tid]), ip + off);
            async_b128(lds_lo32(&s_j[0][tid]), jp + off);
        }
        for (int t = 0; t < ntiles; ++t) {
            const int buf = t & 1;
            if (t + 1 < ntiles) {
                const int    nb  = buf ^ 1;
                const size_t off = (size_t)(tile0 + t + 1) * BLOCK + tid;
                async_b128(lds_lo32(&s_n[nb][tid]), np + off);
                async_b128(lds_lo32(&s_i[nb][tid]), ip + off);
                async_b128(lds_lo32(&s_j[nb][tid]), jp + off);
                // 6 outstanding per wave -> wait until the current tile's 3
                // ops retire (async loads complete in order).
                asm volatile("s_wait_asynccnt 3" ::: "memory");
            } else {
                asm volatile("s_wait_asynccnt 0" ::: "memory");
            }
            // Each lane reads back exactly the 16B slices it loaded: no
            // cross-wave sharing, so no barrier needed in the hot loop.
            const int4 nv = s_n[buf][tid];
            const int4 iv = s_i[buf][tid];
            const int4 jv = s_j[buf][tid];
            do_pair(nv.x, iv.x, jv.x, A, E, pos, elm, s_rs2, mask);
            do_pair(nv.y, iv.y, jv.y, A, E, pos, elm, s_rs2, mask);
            do_pair(nv.z, iv.z, jv.z, A, E, pos, elm, s_rs2, mask);
            do_pair(nv.w, iv.w, jv.w, A, E, pos, elm, s_rs2, mask);
        }
    }

    // Wave32 OR-reduction, then one LDS 64-bit atomicOr per wave.
    #pragma unroll
    for (int o = 16; o > 0; o >>= 1)
        mask |= __shfl_xor(mask, o, 32);
    if ((tid & 31) == 0 && mask) atomicOr(&s_mask, mask);
    __syncthreads();

    // <=32 conditional stores of an identical value: benign across blocks.
    if (tid < B && ((s_mask >> tid) & 1ull)) out[tid] = 1.0f;
}

// ---- Tail kernel (only if P is not a multiple of TILE) --------------------
__global__ void pair_tail_kernel(const float* __restrict__ pos,
                                 const int*   __restrict__ elm,
                                 const float* __restrict__ radius,
                                 const int*   __restrict__ n,
                                 const int*   __restrict__ i,
                                 const int*   __restrict__ j,
                                 float*       __restrict__ out,
                                 int A, int start, int P) {
    const int p = start + blockIdx.x * blockDim.x + threadIdx.x;
    if (p >= P) return;
    const int nn = n[p], ii = i[p], jj = j[p];
    const int base = nn * A;
    const float* pi = pos + (size_t)(base + ii) * 3u;
    const float* pj = pos + (size_t)(base + jj) * 3u;
    const float dx = pj[0] - pi[0];
    const float dy = pj[1] - pi[1];
    const float dz = pj[2] - pi[2];
    const float sod = dx * dx + dy * dy + dz * dz;
    const float rs  = radius[elm[base + ii]] + radius[elm[base + jj]];
    if (rs * rs >= sod) out[nn] = 1.0f;
}

extern "C" void kernel_launch(void* const* d_in, const int* in_sizes, int n_in,
                              void* d_out, int out_size, void* d_ws, size_t ws_size,
                              hipStream_t stream) {
    const float* pos     = (const float*)d_in[0];
    const float* eng     = (const float*)d_in[1];
    const int*   elm     = (const int*)  d_in[2];
    const float* radius  = (const float*)d_in[3];
    const float* eng_atm = (const float*)d_in[4];
    const int*   n       = (const int*)  d_in[5];
    const int*   i       = (const int*)  d_in[6];
    const int*   j       = (const int*)  d_in[7];
    float*       out     = (float*)d_out;

    const int B = in_sizes[1];            // 32
    const int A = in_sizes[2] / B;        // 4096
    const int E = in_sizes[3];            // 8
    const int P = in_sizes[5];            // 8,388,608

    // 1) eng_big + initialize out[] (stream-ordered before pair kernel).
    eng_kernel<<<B, BLOCK, 0, stream>>>(eng, elm, eng_atm, out, A, E);

    // 2) pair contacts via async double-buffered tiles.
    const int numTiles = P / TILE;        // 8192 for the reference sizes
    const int rem      = P - numTiles * TILE;
    if (numTiles > 0) {
        const int blocks = (numTiles + TPB - 1) / TPB;  // 1024
        pair_kernel<<<blocks, BLOCK, 0, stream>>>(pos, elm, radius, n, i, j,
                                                  out, A, E, numTiles, B);
    }
    if (rem > 0) {
        pair_tail_kernel<<<(rem + 255) / 256, 256, 0, stream>>>(
            pos, elm, radius, n, i, j, out, A, numTiles * TILE, P);
    }
}